// GCN_652835029797
// MI455X (gfx1250) — compile-verified
//
#include <hip/hip_runtime.h>

typedef __attribute__((ext_vector_type(2))) float v2f;
typedef __attribute__((ext_vector_type(8))) float v8f;

// ---------------------------------------------------------------------------
// Degree / normalization
// ---------------------------------------------------------------------------
__global__ __launch_bounds__(256) void gcn_deg_init(float* __restrict__ deg, int n) {
  int i = blockIdx.x * 256 + threadIdx.x;
  if (i < n) deg[i] = 1.0f;  // self-loop contributes 1
}

__global__ __launch_bounds__(256) void gcn_deg_accum(const int* __restrict__ dst,
                                                     float* __restrict__ deg, int e) {
  int i = blockIdx.x * 256 + threadIdx.x;
  if (i < e) atomicAdd(&deg[dst[i]], 1.0f);
}

__global__ __launch_bounds__(256) void gcn_dinv(float* __restrict__ degv, int n) {
  int i = blockIdx.x * 256 + threadIdx.x;
  if (i < n) {
    float d = degv[i];
    degv[i] = (d > 0.0f) ? rsqrtf(d) : 0.0f;  // in place: deg -> dinv
  }
}

// per-edge symmetric-norm weight, computed once and reused for all 3 layers
__global__ __launch_bounds__(256) void gcn_edge_norm(const int* __restrict__ src,
                                                     const int* __restrict__ dst,
                                                     const float* __restrict__ dinv,
                                                     float* __restrict__ norm, int e) {
  int i = blockIdx.x * 256 + threadIdx.x;
  if (i < e) norm[i] = dinv[src[i]] * dinv[dst[i]];
}

__global__ __launch_bounds__(256) void gcn_zero4(float4* __restrict__ p, int count4) {
  int i = blockIdx.x * 256 + threadIdx.x;
  if (i < count4) p[i] = make_float4(0.f, 0.f, 0.f, 0.f);
}

// W[kdim x dout] -> Wt[dout x kdim] so each lane's B fragment is a contiguous v2f
__global__ __launch_bounds__(256) void gcn_transpose(const float* __restrict__ W,
                                                     float* __restrict__ Wt,
                                                     int kdim, int dout) {
  int idx = blockIdx.x * 256 + threadIdx.x;
  if (idx >= kdim * dout) return;
  int k = idx / dout;
  int c = idx - k * dout;
  Wt[(size_t)c * kdim + k] = W[idx];
}

// ---------------------------------------------------------------------------
// Dense GEMM: out[n x dout] = H[n x KDIM] @ W[KDIM x dout]  (W pre-transposed)
// One wave owns a 16-wide column strip; B tile (KDIM x 16) lives in registers
// (KDIM/4 v2f = 64 VGPRs for KDIM=128) and is reused across MPW row-tiles.
// f32 WMMA 16x16x4 layouts:
//   A (16x4): lane(h=lane>>4, r=lane&15) holds A[r][2h], A[r][2h+1]
//   B (4x16): VGPR j, half h holds B[2h+j][r]
//   C/D     : VGPR v, half h holds C[v+8h][r]
// ---------------------------------------------------------------------------
template <int KDIM, int MPW>
__global__ __launch_bounds__(256) void gcn_gemm_wmma(
    const float* __restrict__ H, const float* __restrict__ Wt,
    float* __restrict__ out, int ntileN, int dout, int mOff, int totalWaves) {
  const int wave = (blockIdx.x * 256 + (int)threadIdx.x) >> 5;
  if (wave >= totalWaves) return;  // uniform per wave: EXEC stays all-ones
  const int lane = threadIdx.x & 31;
  const int half = lane >> 4;
  const int r    = lane & 15;

  const int group = wave / ntileN;
  const int nTile = wave - group * ntileN;
  const int col   = nTile * 16 + r;

  // Preload this column strip of B into registers.
  v2f breg[KDIM / 4];
  const float* __restrict__ wcol = Wt + (size_t)col * KDIM + 2 * half;
#pragma unroll
  for (int kk = 0; kk < KDIM / 4; ++kk)
    breg[kk] = *(const v2f*)(wcol + kk * 4);

  const int mBase = (mOff + group * MPW) * 16;
#pragma unroll
  for (int m = 0; m < MPW; ++m) {
    const int rowBase = mBase + m * 16;
    const float* __restrict__ hrow = H + (size_t)(rowBase + r) * KDIM + 2 * half;
    v8f acc = {};
#pragma unroll
    for (int kk = 0; kk < KDIM / 4; ++kk) {
      v2f a = *(const v2f*)(hrow + kk * 4);
      acc = __builtin_amdgcn_wmma_f32_16x16x4_f32(
          /*neg_a=*/false, a, /*neg_b=*/false, breg[kk],
          /*c_mod=*/(short)0, acc, /*reuse_a=*/false, /*reuse_b=*/false);
    }
    float* __restrict__ orow = out + (size_t)(rowBase + 8 * half) * dout + col;
#pragma unroll
    for (int v = 0; v < 8; ++v)
      orow[(size_t)v * dout] = acc[v];
  }
}

// ---------------------------------------------------------------------------
// Edge aggregation: agg[dst] += hW[src] * norm[edge]
// One thread per (edge, 4 features): float4 gather + 4 f32 atomics; a wave
// covers one edge's contiguous feature row -> coalesced gathers/atomics and
// broadcast index loads.
// ---------------------------------------------------------------------------
__global__ __launch_bounds__(256) void gcn_edge_agg4(
    const int* __restrict__ src, const int* __restrict__ dst,
    const float* __restrict__ norm, const float* __restrict__ hw,
    float* __restrict__ agg, int e, int qshift /*log2(dout/4)*/, int dout) {
  const int idx = blockIdx.x * 256 + threadIdx.x;
  const int ed  = idx >> qshift;
  if (ed >= e) return;
  const int q = idx & ((1 << qshift) - 1);
  const int s = src[ed];
  const int d = dst[ed];
  const float w = norm[ed];
  const float4 v = *(const float4*)(hw + (size_t)s * dout + q * 4);
  float* a = agg + (size_t)d * dout + q * 4;
  atomicAdd(a + 0, v.x * w);
  atomicAdd(a + 1, v.y * w);
  atomicAdd(a + 2, v.z * w);
  atomicAdd(a + 3, v.w * w);
}

// ---------------------------------------------------------------------------
// Finalize: out = [relu]( agg + hW * dinv^2 (self-loop) + bias ), float4 wide.
// agg and out may alias (elementwise).
// ---------------------------------------------------------------------------
__global__ __launch_bounds__(256) void gcn_finalize4(
    const float* __restrict__ agg, const float* __restrict__ hw,
    const float* __restrict__ dinv, const float* __restrict__ bias,
    float* __restrict__ out, int n, int qshift, int do_relu) {
  const int idx  = blockIdx.x * 256 + threadIdx.x;
  const int node = idx >> qshift;
  if (node >= n) return;
  const int q    = idx & ((1 << qshift) - 1);
  const float di = dinv[node];
  const float w  = di * di;
  const size_t base = (size_t)idx * 4;
  const float4 a = *(const float4*)(agg + base);
  const float4 h = *(const float4*)(hw + base);
  const float4 b = *(const float4*)(bias + q * 4);
  float4 o;
  o.x = a.x + h.x * w + b.x;
  o.y = a.y + h.y * w + b.y;
  o.z = a.z + h.z * w + b.z;
  o.w = a.w + h.w * w + b.w;
  if (do_relu) {
    o.x = fmaxf(o.x, 0.f); o.y = fmaxf(o.y, 0.f);
    o.z = fmaxf(o.z, 0.f); o.w = fmaxf(o.w, 0.f);
  }
  *(float4*)(out + base) = o;
}

// ---------------------------------------------------------------------------
// Launch
// ---------------------------------------------------------------------------
static inline int cdiv_ll(long long a, long long b) { return (int)((a + b - 1) / b); }

static void launch_gemm(const float* H, const float* Wt, float* out,
                        int n, int dout, int kdim, hipStream_t stream) {
  const int MPW = 5;
  const int ntileN = dout / 16;
  const int mtiles = n / 16;
  const int groups = mtiles / MPW;
  const int rem    = mtiles - groups * MPW;
  if (groups > 0) {
    int waves = groups * ntileN;
    gcn_gemm_wmma<128, 5><<<cdiv_ll((long long)waves * 32, 256), dim3(256), 0, stream>>>(
        H, Wt, out, ntileN, dout, 0, waves);
  }
  if (rem > 0) {
    int waves = rem * ntileN;
    gcn_gemm_wmma<128, 1><<<cdiv_ll((long long)waves * 32, 256), dim3(256), 0, stream>>>(
        H, Wt, out, ntileN, dout, groups * MPW, waves);
  }
  (void)kdim;
}

extern "C" void kernel_launch(void* const* d_in, const int* in_sizes, int n_in,
                              void* d_out, int out_size, void* d_ws, size_t ws_size,
                              hipStream_t stream) {
  const float* x  = (const float*)d_in[0];
  const float* W0 = (const float*)d_in[1];
  const float* b0 = (const float*)d_in[2];
  const float* W1 = (const float*)d_in[3];
  const float* b1 = (const float*)d_in[4];
  const float* W2 = (const float*)d_in[5];
  const float* b2 = (const float*)d_in[6];
  const int*  src = (const int*)d_in[7];
  const int*  dst = (const int*)d_in[8];

  const int KIN = 128, DHID = 128, DOUTC = 64;
  const int n = in_sizes[0] / KIN;   // 100000
  const int e = in_sizes[7];         // 1600000

  char* ws = (char*)d_ws;
  size_t off = 0;
  float* dinv = (float*)(ws + off); off += (((size_t)n * 4) + 255) & ~(size_t)255;
  float* norm = (float*)(ws + off); off += (((size_t)e * 4) + 255) & ~(size_t)255;
  float* Wt   = (float*)(ws + off); off += (((size_t)KIN * DHID * 4) + 255) & ~(size_t)255;
  float* buf1 = (float*)(ws + off); off += (((size_t)n * DHID * 4) + 255) & ~(size_t)255;
  float* buf2 = (float*)(ws + off);

  float* out = (float*)d_out;
  dim3 blk(256);

  // --- symmetric normalization: deg = 1 + in-degree, dinv = rsqrt(deg) ---
  gcn_deg_init <<<cdiv_ll(n, 256), blk, 0, stream>>>(dinv, n);
  gcn_deg_accum<<<cdiv_ll(e, 256), blk, 0, stream>>>(dst, dinv, e);
  gcn_dinv     <<<cdiv_ll(n, 256), blk, 0, stream>>>(dinv, n);
  gcn_edge_norm<<<cdiv_ll(e, 256), blk, 0, stream>>>(src, dst, dinv, norm, e);

  // --- layer 0: h1 = relu(agg(x @ W0) + b0) ---
  gcn_transpose<<<cdiv_ll(KIN * DHID, 256), blk, 0, stream>>>(W0, Wt, KIN, DHID);
  launch_gemm(x, Wt, buf1, n, DHID, KIN, stream);
  gcn_zero4    <<<cdiv_ll((long long)n * DHID / 4, 256), blk, 0, stream>>>((float4*)buf2, n * DHID / 4);
  gcn_edge_agg4<<<cdiv_ll((long long)e * (DHID / 4), 256), blk, 0, stream>>>(src, dst, norm, buf1, buf2, e, 5, DHID);
  gcn_finalize4<<<cdiv_ll((long long)n * (DHID / 4), 256), blk, 0, stream>>>(buf2, buf1, dinv, b0, buf2, n, 5, 1);

  // --- layer 1: h2 = relu(agg(h1 @ W1) + b1) ---
  gcn_transpose<<<cdiv_ll(DHID * DHID, 256), blk, 0, stream>>>(W1, Wt, DHID, DHID);
  launch_gemm(buf2, Wt, buf1, n, DHID, DHID, stream);
  gcn_zero4    <<<cdiv_ll((long long)n * DHID / 4, 256), blk, 0, stream>>>((float4*)buf2, n * DHID / 4);
  gcn_edge_agg4<<<cdiv_ll((long long)e * (DHID / 4), 256), blk, 0, stream>>>(src, dst, norm, buf1, buf2, e, 5, DHID);
  gcn_finalize4<<<cdiv_ll((long long)n * (DHID / 4), 256), blk, 0, stream>>>(buf2, buf1, dinv, b1, buf2, n, 5, 1);

  // --- layer 2: out = agg(h2 @ W2) + b2 (no relu) ---
  gcn_transpose<<<cdiv_ll(DHID * DOUTC, 256), blk, 0, stream>>>(W2, Wt, DHID, DOUTC);
  launch_gemm(buf2, Wt, buf1, n, DOUTC, DHID, stream);
  gcn_zero4    <<<cdiv_ll((long long)n * DOUTC / 4, 256), blk, 0, stream>>>((float4*)out, n * DOUTC / 4);
  gcn_edge_agg4<<<cdiv_ll((long long)e * (DOUTC / 4), 256), blk, 0, stream>>>(src, dst, norm, buf1, out, e, 4, DOUTC);
  gcn_finalize4<<<cdiv_ll((long long)n * (DOUTC / 4), 256), blk, 0, stream>>>(out, buf1, dinv, b2, out, n, 4, 0);
}